// ErodeDGNN_10393820856802
// MI455X (gfx1250) — compile-verified
//
#include <hip/hip_runtime.h>
#include <math.h>

#define NPTS 8192
#define KNN  20

typedef __attribute__((ext_vector_type(2))) float v2f;
typedef __attribute__((ext_vector_type(8))) float v8f;

#define CAT_LD 424   // 420 concat cols + 4 zero pad (K padded to mult of 8)

// ---------------------------------------------------------------------------
// Pad x [N,3] -> [N,4] (zero 4th col) and compute row squared norms.
// ---------------------------------------------------------------------------
__global__ void padx_kernel(const float* __restrict__ x,
                            float* __restrict__ f0p,
                            float* __restrict__ sq, int n) {
    int i = blockIdx.x * blockDim.x + threadIdx.x;
    if (i >= n) return;
    float a = x[3 * i + 0], b = x[3 * i + 1], c = x[3 * i + 2];
    f0p[4 * i + 0] = a; f0p[4 * i + 1] = b;
    f0p[4 * i + 2] = c; f0p[4 * i + 3] = 0.0f;
    sq[i] = a * a + b * b + c * c;
}

// Zero the pad columns [420..423] of cat.
__global__ void zeropad_cat_kernel(float* __restrict__ cat, int n) {
    int i = blockIdx.x * blockDim.x + threadIdx.x;
    if (i >= n * 4) return;
    int row = i >> 2, c = 420 + (i & 3);
    cat[(size_t)row * CAT_LD + c] = 0.0f;
}

// ---------------------------------------------------------------------------
// Row squared norms of a strided feature block.
// ---------------------------------------------------------------------------
__global__ void sqnorm_kernel(const float* __restrict__ feat, int ld, int C,
                              float* __restrict__ sq, int n) {
    int i = blockIdx.x * blockDim.x + threadIdx.x;
    if (i >= n) return;
    const float* row = feat + (size_t)i * ld;
    float s = 0.0f;
    for (int c = 0; c < C; ++c) { float v = row[c]; s += v * v; }
    sq[i] = s;
}

// ---------------------------------------------------------------------------
// Weight transpose + zero-pad: Bt[n*Kdp + k] = (n<Nc && k<Kd) ? B[k*Nc+n] : 0
// ---------------------------------------------------------------------------
__global__ void transw_kernel(const float* __restrict__ B, int Kd, int Nc,
                              float* __restrict__ Bt, int Kdp, int Ncp) {
    int i = blockIdx.x * blockDim.x + threadIdx.x;
    if (i >= Kdp * Ncp) return;
    int n = i / Kdp, k = i % Kdp;
    Bt[i] = (n < Nc && k < Kd) ? B[(size_t)k * Nc + n] : 0.0f;
}

// ---------------------------------------------------------------------------
// Fused Gram + top-K-min. One wave owns 16 rows. A fragments (invariant over
// all column tiles) are hoisted into registers; per tile all B fragments are
// loaded first, then the full WMMA chain runs back-to-back (one wait / tile).
// Score s(j) = ||x_j||^2 - 2*x_i.x_j (row-constant term dropped; ordering-
// invariant, self is always minimal, matching jax top_k(-dist)).
// ---------------------------------------------------------------------------
template <int CSTEPS>
__global__ __launch_bounds__(32) void gram_topk_kernel(
    const float* __restrict__ feat, int ld,
    const float* __restrict__ sq, int* __restrict__ idx_out) {
    __shared__ float tile[16 * 16];
    __shared__ float sqs[16];

    const int lane = threadIdx.x;
    const int half = lane >> 4;
    const int l16  = lane & 15;
    const int rowBase = blockIdx.x * 16;

    // Hoisted A fragments: a row of 16 points, K = CSTEPS*4 features.
    const float* aptr = feat + (size_t)(rowBase + l16) * ld + 2 * half;
    v2f afrag[CSTEPS];
#pragma unroll
    for (int s = 0; s < CSTEPS; ++s)
        afrag[s] = *(const v2f*)(aptr + 4 * s);

    float vals[KNN];
    int   inds[KNN];
#pragma unroll
    for (int t = 0; t < KNN; ++t) { vals[t] = 3.0e38f; inds[t] = 0; }

    for (int ct = 0; ct < NPTS / 16; ++ct) {
        const int colBase = ct * 16;
        const float* bptr = feat + (size_t)(colBase + l16) * ld + 2 * half;

        v2f bfrag[CSTEPS];
#pragma unroll
        for (int s = 0; s < CSTEPS; ++s)
            bfrag[s] = *(const v2f*)(bptr + 4 * s);

        v8f acc = {};
#pragma unroll
        for (int s = 0; s < CSTEPS; ++s)
            acc = __builtin_amdgcn_wmma_f32_16x16x4_f32(
                false, afrag[s], false, bfrag[s], (short)0, acc, false, false);

        __syncthreads();   // previous scan done before tile overwrite
#pragma unroll
        for (int r = 0; r < 8; ++r)
            tile[(r + 8 * half) * 16 + l16] = acc[r];
        if (half == 0) sqs[l16] = sq[colBase + l16];
        __syncthreads();

        if (lane < 16) {
            const int r = lane;
#pragma unroll
            for (int c = 0; c < 16; ++c) {
                float d = sqs[c] - 2.0f * tile[r * 16 + c];
                if (d < vals[KNN - 1]) {
                    float v = d; int id = colBase + c;
#pragma unroll
                    for (int t = 0; t < KNN; ++t) {
                        if (v < vals[t]) {
                            float tv = vals[t]; int ti = inds[t];
                            vals[t] = v; inds[t] = id; v = tv; id = ti;
                        }
                    }
                }
            }
        }
    }

    if (lane < 16) {
#pragma unroll
        for (int t = 0; t < KNN; ++t)
            idx_out[(size_t)(rowBase + lane) * KNN + t] = inds[t];
    }
}

// ---------------------------------------------------------------------------
// Morphological erosion EdgeConv: out[i, f*Cin+c] = min_k fin[nbr_k, c] - w[f,k,c]
// ---------------------------------------------------------------------------
__global__ void erode_kernel(const float* __restrict__ fin, int ldin, int Cin,
                             const float* __restrict__ w, int F,
                             const int* __restrict__ idx,
                             float* __restrict__ out, int ldo, int colOff) {
    __shared__ int nb[KNN];
    const int i = blockIdx.x;
    const int tid = threadIdx.x;
    if (tid < KNN) nb[tid] = idx[(size_t)i * KNN + tid];
    __syncthreads();
    const int FC = F * Cin;
    if (tid < FC) {
        const int f = tid / Cin, c = tid % Cin;
        float m = 3.0e38f;
#pragma unroll
        for (int k = 0; k < KNN; ++k) {
            float v = fin[(size_t)nb[k] * ldin + c] - w[(f * KNN + k) * Cin + c];
            m = fminf(m, v);
        }
        out[(size_t)i * ldo + colOff + tid] = m;
    }
}

// ---------------------------------------------------------------------------
// WMMA f32 GEMM vs pre-transposed zero-padded weights Bt[Ncp][Kdp]:
// Out[M,Nc] = act(A[M,Kdp] @ B + bias). Wave computes a 32x64 tile.
// Kdp % 8 == 0 -> per iteration: 12 b64 loads, then 16 back-to-back WMMAs.
// ---------------------------------------------------------------------------
__global__ __launch_bounds__(32) void gemm_bias_act_kernel(
    const float* __restrict__ A, int lda,
    const float* __restrict__ Bt, int Kdp, int Nc,
    const float* __restrict__ bias,
    float* __restrict__ Out, int ldo, int relu) {
    const int lane = threadIdx.x;
    const int half = lane >> 4;
    const int l16  = lane & 15;
    const int rowBase = blockIdx.y * 32;
    const int colBase = blockIdx.x * 64;

    v8f acc[2][4];
#pragma unroll
    for (int mi = 0; mi < 2; ++mi)
#pragma unroll
        for (int ni = 0; ni < 4; ++ni) acc[mi][ni] = {};

    const float* aptr0 = A + (size_t)(rowBase + l16) * lda + 2 * half;
    const float* aptr1 = aptr0 + (size_t)16 * lda;
    const float* btp[4];
#pragma unroll
    for (int ni = 0; ni < 4; ++ni)
        btp[ni] = Bt + (size_t)(colBase + ni * 16 + l16) * Kdp + 2 * half;

    for (int kk = 0; kk < Kdp; kk += 8) {
        v2f a0[2], a1[2], b[4][2];
        a0[0] = *(const v2f*)(aptr0 + kk);
        a0[1] = *(const v2f*)(aptr0 + kk + 4);
        a1[0] = *(const v2f*)(aptr1 + kk);
        a1[1] = *(const v2f*)(aptr1 + kk + 4);
#pragma unroll
        for (int ni = 0; ni < 4; ++ni) {
            b[ni][0] = *(const v2f*)(btp[ni] + kk);
            b[ni][1] = *(const v2f*)(btp[ni] + kk + 4);
        }
#pragma unroll
        for (int u = 0; u < 2; ++u) {
#pragma unroll
            for (int ni = 0; ni < 4; ++ni) {
                acc[0][ni] = __builtin_amdgcn_wmma_f32_16x16x4_f32(
                    false, a0[u], false, b[ni][u], (short)0, acc[0][ni], false, false);
                acc[1][ni] = __builtin_amdgcn_wmma_f32_16x16x4_f32(
                    false, a1[u], false, b[ni][u], (short)0, acc[1][ni], false, false);
            }
        }
    }

#pragma unroll
    for (int ni = 0; ni < 4; ++ni) {
        const int col = colBase + ni * 16 + l16;
        if (col >= Nc) continue;
        const float bv = bias[col];
#pragma unroll
        for (int mi = 0; mi < 2; ++mi) {
#pragma unroll
            for (int r = 0; r < 8; ++r) {
                const int row = rowBase + mi * 16 + r + 8 * half;
                float v = acc[mi][ni][r] + bv;
                if (relu) v = fmaxf(v, 0.0f);
                Out[(size_t)row * ldo + col] = v;
            }
        }
    }
}

// ---------------------------------------------------------------------------
// Row-wise log_softmax, one thread per row (C=40).
// ---------------------------------------------------------------------------
__global__ void logsoftmax_kernel(const float* __restrict__ z,
                                  float* __restrict__ out, int n, int C) {
    int i = blockIdx.x * blockDim.x + threadIdx.x;
    if (i >= n) return;
    const float* zi = z + (size_t)i * C;
    float mx = -3.0e38f;
    for (int c = 0; c < C; ++c) mx = fmaxf(mx, zi[c]);
    float s = 0.0f;
    for (int c = 0; c < C; ++c) s += __expf(zi[c] - mx);
    float l = mx + __logf(s);
    float* oi = out + (size_t)i * C;
    for (int c = 0; c < C; ++c) oi[c] = zi[c] - l;
}

// ---------------------------------------------------------------------------
// Orchestration
// ---------------------------------------------------------------------------
extern "C" void kernel_launch(void* const* d_in, const int* in_sizes, int n_in,
                              void* d_out, int out_size, void* d_ws, size_t ws_size,
                              hipStream_t stream) {
    const int N = NPTS;
    const float* x      = (const float*)d_in[0];
    const float* w1     = (const float*)d_in[1];
    const float* w2     = (const float*)d_in[2];
    const float* w3     = (const float*)d_in[3];
    const float* lin1_w = (const float*)d_in[4];
    const float* lin1_b = (const float*)d_in[5];
    const float* wa     = (const float*)d_in[6];
    const float* ba     = (const float*)d_in[7];
    const float* wb     = (const float*)d_in[8];
    const float* bb     = (const float*)d_in[9];
    const float* wo     = (const float*)d_in[10];
    const float* bo     = (const float*)d_in[11];

    float* ws  = (float*)d_ws;
    float* f0p = ws;                             // N*4
    float* sq  = f0p + (size_t)N * 4;            // N
    float* cat = sq  + (size_t)N;                // N*424 : [x1 | x2 | x3 | 0pad]
    float* h1  = cat + (size_t)N * CAT_LD;       // N*1024
    float* h2  = h1  + (size_t)N * 1024;         // N*256
    float* h3  = h2  + (size_t)N * 256;          // N*128
    float* h4  = h3  + (size_t)N * 128;          // N*40
    float* bt1 = h4  + (size_t)N * 40;           // 1024*424
    float* bt2 = bt1 + (size_t)1024 * 424;       // 256*1024
    float* bt3 = bt2 + (size_t)256 * 1024;       // 128*256
    float* bt4 = bt3 + (size_t)128 * 256;        // 64*128
    int*   idx = (int*)(bt4 + (size_t)64 * 128); // N*20

    // Weight transposes (+ zero pad) for the MLP; independent of point data.
    transw_kernel<<<(1024 * 424 + 255) / 256, 256, 0, stream>>>(lin1_w, 420, 1024, bt1, 424, 1024);
    transw_kernel<<<(256 * 1024 + 255) / 256, 256, 0, stream>>>(wa, 1024, 256, bt2, 1024, 256);
    transw_kernel<<<(128 * 256 + 255) / 256, 256, 0, stream>>>(wb, 256, 128, bt3, 256, 128);
    transw_kernel<<<(64 * 128 + 255) / 256, 256, 0, stream>>>(wo, 128, 40, bt4, 128, 64);
    zeropad_cat_kernel<<<(N * 4 + 255) / 256, 256, 0, stream>>>(cat, N);

    // ---- layer 1: knn on x (C=3, padded to 4), erode -> cat[:, 0:60]
    padx_kernel<<<N / 256, 256, 0, stream>>>(x, f0p, sq, N);
    gram_topk_kernel<1><<<N / 16, 32, 0, stream>>>(f0p, 4, sq, idx);
    erode_kernel<<<N, 64, 0, stream>>>(x, 3, 3, w1, 20, idx, cat, CAT_LD, 0);

    // ---- layer 2: knn on x1 (C=60), erode -> cat[:, 60:180]
    sqnorm_kernel<<<N / 256, 256, 0, stream>>>(cat, CAT_LD, 60, sq, N);
    gram_topk_kernel<15><<<N / 16, 32, 0, stream>>>(cat, CAT_LD, sq, idx);
    erode_kernel<<<N, 128, 0, stream>>>(cat, CAT_LD, 60, w2, 2, idx, cat, CAT_LD, 60);

    // ---- layer 3: knn on x2 (C=120), erode -> cat[:, 180:420]
    sqnorm_kernel<<<N / 256, 256, 0, stream>>>(cat + 60, CAT_LD, 120, sq, N);
    gram_topk_kernel<30><<<N / 16, 32, 0, stream>>>(cat + 60, CAT_LD, sq, idx);
    erode_kernel<<<N, 256, 0, stream>>>(cat + 60, CAT_LD, 120, w3, 2, idx, cat, CAT_LD, 180);

    // ---- MLP head (A @ Bt^T with Bt pre-transposed, K and N zero-padded)
    dim3 g1(1024 / 64, N / 32);
    gemm_bias_act_kernel<<<g1, 32, 0, stream>>>(cat, CAT_LD, bt1, 424, 1024, lin1_b, h1, 1024, 1);
    dim3 g2(256 / 64, N / 32);
    gemm_bias_act_kernel<<<g2, 32, 0, stream>>>(h1, 1024, bt2, 1024, 256, ba, h2, 256, 1);
    dim3 g3(128 / 64, N / 32);
    gemm_bias_act_kernel<<<g3, 32, 0, stream>>>(h2, 256, bt3, 256, 128, bb, h3, 128, 1);
    dim3 g4(64 / 64, N / 32);
    gemm_bias_act_kernel<<<g4, 32, 0, stream>>>(h3, 128, bt4, 128, 40, bo, h4, 40, 0);

    logsoftmax_kernel<<<N / 256, 256, 0, stream>>>(h4, (float*)d_out, N, 40);
}